// GAT_82351702933925
// MI455X (gfx1250) — compile-verified
//
#include <hip/hip_runtime.h>

// ---------------- problem constants (match reference) ----------------
#define Nn        50000
#define IN_DIMc   512
#define HEADSc    8
#define HIDc      8
#define HHc       64          // HEADS*HID
#define OUTc      7
#define Ee        1600000
#define NEG_SLOPE 0.2f
#define TOTE      (Ee + Nn)   // edges + self loops

typedef __attribute__((ext_vector_type(2))) float v2f;
typedef __attribute__((ext_vector_type(8))) float v8f;

// ---------------- workspace layout (floats) ----------------
#define OFF_H1    ((size_t)0)                          // [N,64]  h1 = x@W1
#define OFF_ASRC1 (OFF_H1    + (size_t)Nn*HHc)         // [N,8]
#define OFF_ADST1 (OFF_ASRC1 + (size_t)Nn*HEADSc)      // [N,8]
#define OFF_MAX1  (OFF_ADST1 + (size_t)Nn*HEADSc)      // [N,8] ordered-uint
#define OFF_SUM1  (OFF_MAX1  + (size_t)Nn*HEADSc)      // [N,8]
#define OFF_ACC1  (OFF_SUM1  + (size_t)Nn*HEADSc)      // [N,64] aggregate -> elu feat
#define OFF_H2    (OFF_ACC1  + (size_t)Nn*HHc)         // [N,16] padded h2
#define OFF_ASRC2 (OFF_H2    + (size_t)Nn*16)          // [N]
#define OFF_ADST2 (OFF_ASRC2 + (size_t)Nn)             // [N]
#define OFF_MAX2  (OFF_ADST2 + (size_t)Nn)             // [N] ordered-uint
#define OFF_SUM2  (OFF_MAX2  + (size_t)Nn)             // [N]
#define WS_FLOATS (OFF_SUM2  + (size_t)Nn)             // ~9.0M floats = 36 MB

// order-preserving float<->uint mapping for atomic max over signed floats
__device__ __forceinline__ unsigned f2ord(float f) {
  unsigned u = __float_as_uint(f);
  return (u & 0x80000000u) ? ~u : (u | 0x80000000u);
}
__device__ __forceinline__ float ord2f(unsigned u) {
  return (u & 0x80000000u) ? __uint_as_float(u & 0x7fffffffu) : __uint_as_float(~u);
}
__device__ __forceinline__ float leaky(float v) { return v > 0.f ? v : v * NEG_SLOPE; }

// ---------------- init: zero accumulators / max bufs / out ----------------
__global__ void init_kernel(float* __restrict__ ws, float* __restrict__ out) {
  size_t stride = (size_t)gridDim.x * blockDim.x;
  size_t t0 = (size_t)blockIdx.x * blockDim.x + threadIdx.x;
  for (size_t i = t0; i < WS_FLOATS; i += stride) ws[i] = 0.f;          // 0u is also min ordered-uint
  for (size_t i = t0; i < (size_t)Nn * OUTc; i += stride) out[i] = 0.f;
}

// ---------------- GEMM1: h1 = x @ W1  via V_WMMA_F32_16X16X4_F32 ----------------
// block = 4 waves; each block owns one 16-row tile of x (staged in LDS, shared
// by all 4 waves); wave w computes column tile [16w, 16w+16) of the 64 outputs.
#define LDS_STRIDE 516   // 512 + 4 pad -> conflict-free (bank = (4r + k) % 64)
__global__ __launch_bounds__(128) void gemm1_wmma(const float* __restrict__ X,
                                                  const float* __restrict__ W,
                                                  float* __restrict__ H1) {
  __shared__ float lds[16 * LDS_STRIDE];
  const int wave = threadIdx.x >> 5;
  const int lane = threadIdx.x & 31;
  const int mt   = blockIdx.x;                 // 0..3124 (50000/16 exact)

  // cooperative, coalesced b128 fill of the shared A tile (16 x 512)
  const float4* xv = (const float4*)(X + (size_t)mt * 16 * IN_DIMc);
  for (int i = threadIdx.x; i < 16 * IN_DIMc / 4; i += 128) {
    int r = i >> 7, k4 = i & 127;              // 128 float4 per row
    ((float4*)(lds + r * LDS_STRIDE))[k4] = xv[i];
  }
  __syncthreads();

  const int mrow = lane & 15;                  // A row / B col within tile
  const int koff = (lane >= 16) ? 2 : 0;       // f32 WMMA K split across lane halves
  const int col  = wave * 16 + mrow;
  const float* arow = lds + mrow * LDS_STRIDE;

  v8f acc = {};
  for (int k = 0; k < IN_DIMc; k += 4) {
    v2f a, b;
    a.x = arow[k + koff];
    a.y = arow[k + koff + 1];
    b.x = W[(size_t)(k + koff)     * HHc + col];
    b.y = W[(size_t)(k + koff + 1) * HHc + col];
    acc = __builtin_amdgcn_wmma_f32_16x16x4_f32(false, a, false, b, (short)0, acc,
                                                false, false);
  }
  const int m0 = mt * 16 + ((lane >= 16) ? 8 : 0);
#pragma unroll
  for (int v = 0; v < 8; ++v)
    H1[(size_t)(m0 + v) * HHc + col] = acc[v];
}

// ---------------- per-node attention scores, layer 1 ----------------
__global__ void att1_kernel(const float* __restrict__ H1,
                            const float* __restrict__ as, const float* __restrict__ ad,
                            float* __restrict__ asrc, float* __restrict__ adst) {
  int t = blockIdx.x * blockDim.x + threadIdx.x;
  if (t >= Nn * HEADSc) return;
  int n = t >> 3, h = t & 7;
  const float* hp = H1 + (size_t)n * HHc + h * HIDc;
  float s = 0.f, d = 0.f;
#pragma unroll
  for (int c = 0; c < HIDc; ++c) { float v = hp[c]; s += v * as[h*HIDc + c]; d += v * ad[h*HIDc + c]; }
  asrc[t] = s; adst[t] = d;
}

// ---------------- edge passes (segment max / sum / aggregate) ----------------
template <int HD>
__global__ void edge_max_kernel(const long long* __restrict__ ei,
                                const float* __restrict__ asrc, const float* __restrict__ adst,
                                unsigned* __restrict__ maxb) {
  int t = blockIdx.x * blockDim.x + threadIdx.x;
  if (t >= TOTE * HD) return;
  int e = t / HD, h = t - e * HD;
  int s, d;
  if (e < Ee) { s = (int)ei[e]; d = (int)ei[(size_t)Ee + e]; } else { s = d = e - Ee; }
  float v = leaky(asrc[(size_t)s * HD + h] + adst[(size_t)d * HD + h]);
  atomicMax(maxb + (size_t)d * HD + h, f2ord(v));
}

template <int HD>
__global__ void edge_sum_kernel(const long long* __restrict__ ei,
                                const float* __restrict__ asrc, const float* __restrict__ adst,
                                const unsigned* __restrict__ maxb, float* __restrict__ sumb) {
  int t = blockIdx.x * blockDim.x + threadIdx.x;
  if (t >= TOTE * HD) return;
  int e = t / HD, h = t - e * HD;
  int s, d;
  if (e < Ee) { s = (int)ei[e]; d = (int)ei[(size_t)Ee + e]; } else { s = d = e - Ee; }
  float v = leaky(asrc[(size_t)s * HD + h] + adst[(size_t)d * HD + h]);
  float m = ord2f(maxb[(size_t)d * HD + h]);
  atomicAdd(sumb + (size_t)d * HD + h, expf(v - m));
}

template <int HD, int C, int FS, int OS>
__global__ void edge_aggr_kernel(const long long* __restrict__ ei,
                                 const float* __restrict__ asrc, const float* __restrict__ adst,
                                 const unsigned* __restrict__ maxb, const float* __restrict__ sumb,
                                 const float* __restrict__ feat, float* __restrict__ out) {
  int t = blockIdx.x * blockDim.x + threadIdx.x;
  if (t >= TOTE * HD) return;
  int e = t / HD, h = t - e * HD;
  int s, d;
  if (e < Ee) { s = (int)ei[e]; d = (int)ei[(size_t)Ee + e]; } else { s = d = e - Ee; }
  float v = leaky(asrc[(size_t)s * HD + h] + adst[(size_t)d * HD + h]);
  float m = ord2f(maxb[(size_t)d * HD + h]);
  float alpha = expf(v - m) / fmaxf(sumb[(size_t)d * HD + h], 1e-16f);
  const float* fp = feat + (size_t)s * FS + h * C;
  float* op = out + (size_t)d * OS + h * C;
#pragma unroll
  for (int c = 0; c < C; ++c) atomicAdd(op + c, alpha * fp[c]);   // L2-resident f32 atomics
}

// ---------------- bias + ELU after layer-1 aggregation ----------------
__global__ void elu_bias_kernel(float* __restrict__ acc, const float* __restrict__ b1) {
  int t = blockIdx.x * blockDim.x + threadIdx.x;
  if (t >= Nn * HHc) return;
  float v = acc[t] + b1[t & (HHc - 1)];
  acc[t] = v > 0.f ? v : expf(v) - 1.f;
}

// ---------------- GEMM2: h2 = elu_feat @ W2 (7 cols padded to 16) ----------------
__global__ __launch_bounds__(128) void gemm2_wmma(const float* __restrict__ Hf,
                                                  const float* __restrict__ W2,
                                                  float* __restrict__ H2) {
  const int wave = threadIdx.x >> 5;
  const int lane = threadIdx.x & 31;
  const int mt = blockIdx.x * 4 + wave;
  if (mt >= Nn / 16) return;                    // wave-uniform exit, no barriers used
  const int mrow = lane & 15;
  const int koff = (lane >= 16) ? 2 : 0;
  const float* arow = Hf + (size_t)(mt * 16 + mrow) * HHc;
  v8f acc = {};
  for (int k = 0; k < HHc; k += 4) {
    v2f a, b;
    a.x = arow[k + koff];
    a.y = arow[k + koff + 1];
    b.x = (mrow < OUTc) ? W2[(size_t)(k + koff)     * OUTc + mrow] : 0.f;
    b.y = (mrow < OUTc) ? W2[(size_t)(k + koff + 1) * OUTc + mrow] : 0.f;
    acc = __builtin_amdgcn_wmma_f32_16x16x4_f32(false, a, false, b, (short)0, acc,
                                                false, false);
  }
  const int m0 = mt * 16 + ((lane >= 16) ? 8 : 0);
#pragma unroll
  for (int v = 0; v < 8; ++v)
    H2[(size_t)(m0 + v) * 16 + mrow] = acc[v];
}

// ---------------- per-node attention scores, layer 2 (1 head) ----------------
__global__ void att2_kernel(const float* __restrict__ H2,
                            const float* __restrict__ as, const float* __restrict__ ad,
                            float* __restrict__ asrc, float* __restrict__ adst) {
  int n = blockIdx.x * blockDim.x + threadIdx.x;
  if (n >= Nn) return;
  const float* hp = H2 + (size_t)n * 16;
  float s = 0.f, d = 0.f;
#pragma unroll
  for (int c = 0; c < OUTc; ++c) { float v = hp[c]; s += v * as[c]; d += v * ad[c]; }
  asrc[n] = s; adst[n] = d;
}

__global__ void final_bias_kernel(float* __restrict__ out, const float* __restrict__ b2) {
  int t = blockIdx.x * blockDim.x + threadIdx.x;
  if (t >= Nn * OUTc) return;
  out[t] += b2[t % OUTc];
}

// ---------------- launch ----------------
extern "C" void kernel_launch(void* const* d_in, const int* in_sizes, int n_in,
                              void* d_out, int out_size, void* d_ws, size_t ws_size,
                              hipStream_t stream) {
  const float*     x   = (const float*)d_in[0];
  const long long* ei  = (const long long*)d_in[1];   // int64 [2, E]
  const float*     W1  = (const float*)d_in[2];
  const float*     as1 = (const float*)d_in[3];
  const float*     ad1 = (const float*)d_in[4];
  const float*     b1  = (const float*)d_in[5];
  const float*     W2  = (const float*)d_in[6];
  const float*     as2 = (const float*)d_in[7];
  const float*     ad2 = (const float*)d_in[8];
  const float*     b2  = (const float*)d_in[9];
  float* out = (float*)d_out;
  float* ws  = (float*)d_ws;

  float*    H1    = ws + OFF_H1;
  float*    ASRC1 = ws + OFF_ASRC1;
  float*    ADST1 = ws + OFF_ADST1;
  unsigned* MAX1  = (unsigned*)(ws + OFF_MAX1);
  float*    SUM1  = ws + OFF_SUM1;
  float*    ACC1  = ws + OFF_ACC1;
  float*    H2    = ws + OFF_H2;
  float*    ASRC2 = ws + OFF_ASRC2;
  float*    ADST2 = ws + OFF_ADST2;
  unsigned* MAX2  = (unsigned*)(ws + OFF_MAX2);
  float*    SUM2  = ws + OFF_SUM2;

  const int TPB = 256;
  const int L1W = TOTE * HEADSc;

  init_kernel<<<4096, TPB, 0, stream>>>(ws, out);

  // ---- layer 1 ----
  gemm1_wmma<<<Nn / 16, 128, 0, stream>>>(x, W1, H1);
  att1_kernel<<<(Nn * HEADSc + TPB - 1) / TPB, TPB, 0, stream>>>(H1, as1, ad1, ASRC1, ADST1);
  edge_max_kernel<HEADSc><<<(L1W + TPB - 1) / TPB, TPB, 0, stream>>>(ei, ASRC1, ADST1, MAX1);
  edge_sum_kernel<HEADSc><<<(L1W + TPB - 1) / TPB, TPB, 0, stream>>>(ei, ASRC1, ADST1, MAX1, SUM1);
  edge_aggr_kernel<HEADSc, HIDc, HHc, HHc>
      <<<(L1W + TPB - 1) / TPB, TPB, 0, stream>>>(ei, ASRC1, ADST1, MAX1, SUM1, H1, ACC1);
  elu_bias_kernel<<<(Nn * HHc + TPB - 1) / TPB, TPB, 0, stream>>>(ACC1, b1);

  // ---- layer 2 ----
  gemm2_wmma<<<(Nn / 16 + 3) / 4, 128, 0, stream>>>(ACC1, W2, H2);
  att2_kernel<<<(Nn + TPB - 1) / TPB, TPB, 0, stream>>>(H2, as2, ad2, ASRC2, ADST2);
  edge_max_kernel<1><<<(TOTE + TPB - 1) / TPB, TPB, 0, stream>>>(ei, ASRC2, ADST2, MAX2);
  edge_sum_kernel<1><<<(TOTE + TPB - 1) / TPB, TPB, 0, stream>>>(ei, ASRC2, ADST2, MAX2, SUM2);
  edge_aggr_kernel<1, OUTc, 16, OUTc>
      <<<(TOTE + TPB - 1) / TPB, TPB, 0, stream>>>(ei, ASRC2, ADST2, MAX2, SUM2, H2, out);
  final_bias_kernel<<<(Nn * OUTc + TPB - 1) / TPB, TPB, 0, stream>>>(out, b2);
}